// HebbianAbcdNN_20624432955929
// MI455X (gfx1250) — compile-verified
//
#include <hip/hip_runtime.h>
#include <hip/hip_bf16.h>
#include <math.h>
#include <stdint.h>

typedef __attribute__((ext_vector_type(2))) float v2f;
typedef __attribute__((ext_vector_type(4))) float v4f;
typedef __attribute__((ext_vector_type(8))) float v8f;
typedef __attribute__((ext_vector_type(4))) unsigned int u32x4;
typedef __attribute__((ext_vector_type(8))) int i32x8;
typedef __attribute__((ext_vector_type(4))) int i32x4;

#define N4K 4096
#define TK 64
#define KSPLIT 4
#define KSEG (N4K / KSPLIT)          // 1024
#define WROWSTRIDE 68                 // 64 + 4 pad floats -> conflict-free A reads
#define TILEF (16 * WROWSTRIDE)       // floats per staged tile (incl. padding)

#if defined(__AMDGCN__) && defined(__has_builtin)
#  if __has_builtin(__builtin_amdgcn_tensor_load_to_lds)
#    define USE_TDM 1
#  else
#    define USE_TDM 0
#  endif
#else
#  define USE_TDM 0
#endif

#if USE_TDM
// ---------------------------------------------------------------------------
// Issue one TDM tile load: 16 rows x 64 floats, global row stride 4096 floats,
// into LDS at lds_byte_off with +4 dwords of padding after every 64 dwords
// (pad_interval=5 -> 64 dw, pad_amount=3 -> 4 dw) => 68-float LDS row stride.
// D# group0: count=1 | lds_addr | global_addr | type=2 ("image")
// D# group1: data_size=2(4B), pad_enable, tensor dims huge (no OOB),
//            tile_dim0=64, tile_dim1=16, tensor_dim0_stride=4096
// ---------------------------------------------------------------------------
__device__ __forceinline__ void tdm_issue_tile(const float* gsrc,
                                               unsigned lds_byte_off)
{
    const unsigned long long ga = (unsigned long long)(uintptr_t)gsrc;
    u32x4 g0;
    g0[0] = 1u;                                        // count=1, user desc
    g0[1] = lds_byte_off;                              // lds_addr (bytes)
    g0[2] = (unsigned)ga;                              // global_addr[31:0]
    g0[3] = (unsigned)((ga >> 32) & 0x01ffffffu)       // global_addr[56:32]
          | (2u << 30);                                // type=2
    i32x8 g1;
    g1[0] = (int)((2u << 16)                           // data_size = 4B
                | (1u << 20)                           // pad_enable
                | (5u << 22)                           // pad_interval: 64 dw
                | (3u << 25));                         // pad_amount: 4 dw
    g1[1] = 0;                                         // abar=0, tdim0[15:0]=0
    g1[2] = (int)0x10u;                                // tdim0=1<<20, tdim1[15:0]=0
    g1[3] = (int)(0x10u | (64u << 16));                // tdim1=1<<20, tile_dim0=64
    g1[4] = 16;                                        // tile_dim1=16, tile_dim2=0
    g1[5] = 4096;                                      // tensor_dim0_stride lo32
    g1[6] = 0;                                         // stride0 hi, stride1 lo
    g1[7] = 0;                                         // stride1 hi
    const i32x4 z4 = {0, 0, 0, 0};
    asm volatile("" ::: "memory");                     // order vs prior LDS reads
#if __clang_major__ >= 23
    const i32x8 z8 = {0, 0, 0, 0, 0, 0, 0, 0};
    __builtin_amdgcn_tensor_load_to_lds(g0, g1, z4, z4, z8, 0);
#else
    __builtin_amdgcn_tensor_load_to_lds(g0, g1, z4, z4, 0);
#endif
}
#endif // USE_TDM

// ---------------------------------------------------------------------------
// GEMV partial: one wave owns 16 output rows over one K segment (1024 cols).
// D(16x16) = A(16x4) x B(4x16) + C, f32 WMMA:
//   A = W tile (lane l: row = l&15, K = 2*(l>>4)+{0,1})        [documented]
//   B = x broadcast over N (lane l, vgpr v: K = v + 2*(l>>4))  [C/D-style striping]
//   => D[m][n] = sum_k W[row0+m][k]*x[k]  (identical across n)
// Block = 256 threads = 8 waves = 128 rows; grid = 32 row-blocks * KSPLIT.
// TDM path: per-wave double-buffered tensor DMA, s_wait_tensorcnt pipelining.
// ---------------------------------------------------------------------------
__global__ __launch_bounds__(256) void gemv_partial_f32wmma(
    const float* __restrict__ W, const float* __restrict__ xsrc,
    float* __restrict__ partial /* [KSPLIT][4096] */)
{
    __shared__ float lds_x[KSEG];
#if USE_TDM
    __shared__ float lds_w[8 * 2 * TILEF];
#else
    __shared__ float lds_w[8 * TILEF];
#endif

    const int tid  = threadIdx.x;
    const int wave = tid >> 5;
    const int lane = tid & 31;
    const int rb   = blockIdx.x & 31;   // row-block 0..31
    const int ks   = blockIdx.x >> 5;   // k-segment 0..KSPLIT-1
    const int kbase = ks * KSEG;
    const int row0  = rb * 128 + wave * 16;

    // stage this segment's x slice (1024 floats = 256 float4, one per thread)
    ((v4f*)lds_x)[tid] = *((const v4f*)(xsrc + kbase) + tid);
    __syncthreads();

    v8f c = {0.f, 0.f, 0.f, 0.f, 0.f, 0.f, 0.f, 0.f};
    const int m  = lane & 15;
    const int kh = (lane >> 4) << 1;        // 0 or 2

#if USE_TDM
    const int ntiles = KSEG / TK;           // 16
    const float* gtile0 = W + (size_t)row0 * N4K + kbase;
    float* buf0 = &lds_w[wave * (2 * TILEF)];
    float* buf1 = buf0 + TILEF;
    const unsigned off0 = (unsigned)(uintptr_t)buf0;   // low 32 bits = LDS offset
    const unsigned off1 = (unsigned)(uintptr_t)buf1;

    tdm_issue_tile(gtile0, off0);
    for (int i = 0; i < ntiles; ++i) {
        if (i + 1 < ntiles) {
            tdm_issue_tile(gtile0 + (i + 1) * TK, ((i + 1) & 1) ? off1 : off0);
            __builtin_amdgcn_s_wait_tensorcnt((short)1);   // oldest (tile i) done
        } else {
            __builtin_amdgcn_s_wait_tensorcnt((short)0);   // last tile done
        }
        asm volatile("" ::: "memory");                     // LDS now holds tile i
        const float* wbuf = (i & 1) ? buf1 : buf0;
        const int k0 = i * TK;
        #pragma unroll
        for (int kk = 0; kk < TK; kk += 4) {
            v2f a = *(const v2f*)&wbuf[m * WROWSTRIDE + kk + kh];
            v2f bb;
            bb[0] = lds_x[k0 + kk + kh];
            bb[1] = lds_x[k0 + kk + kh + 1];
            c = __builtin_amdgcn_wmma_f32_16x16x4_f32(
                    false, a, false, bb, (short)0, c, false, false);
        }
    }
#else
    const int wbase = wave * TILEF;
    for (int k0 = 0; k0 < KSEG; k0 += TK) {
        __syncthreads();   // WAR: previous tile fully consumed
        #pragma unroll
        for (int t = lane; t < 256; t += 32) {
            const int r  = t >> 4;
            const int cc = (t & 15) << 2;
            v4f v = __builtin_nontemporal_load(
                (const v4f*)(W + (size_t)(row0 + r) * N4K + kbase + k0 + cc));
            *(v4f*)&lds_w[wbase + r * WROWSTRIDE + cc] = v;
        }
        __syncthreads();   // RAW: tile visible to all lanes
        #pragma unroll
        for (int kk = 0; kk < TK; kk += 4) {
            v2f a = *(const v2f*)&lds_w[wbase + m * WROWSTRIDE + kk + kh];
            v2f bb;
            bb[0] = lds_x[k0 + kk + kh];
            bb[1] = lds_x[k0 + kk + kh + 1];
            c = __builtin_amdgcn_wmma_f32_16x16x4_f32(
                    false, a, false, bb, (short)0, c, false, false);
        }
    }
#endif

    // C/D layout: vgpr j -> row j (lanes 0-15) / row j+8 (lanes 16-31)
    if ((lane & 15) == 0) {
        const int rbase = row0 + (lane >> 4) * 8;
        #pragma unroll
        for (int j = 0; j < 8; ++j)
            partial[ks * N4K + rbase + j] = c[j];
    }
}

// ---------------------------------------------------------------------------
// Reduce KSPLIT partials + bias + activation.  mode 0: tanh -> post
// mode 1: raw -> yraw, sigmoid -> post
// ---------------------------------------------------------------------------
__global__ __launch_bounds__(256) void act_bias(
    const float* __restrict__ partial, const float* __restrict__ b,
    float* __restrict__ post, float* __restrict__ yraw, int mode)
{
    const int r = blockIdx.x * 256 + threadIdx.x;   // 4096 threads total
    float s = b[r];
    #pragma unroll
    for (int ks = 0; ks < KSPLIT; ++ks) s += partial[ks * N4K + r];
    if (mode == 0) {
        post[r] = tanhf(s);
    } else {
        yraw[r] = s;
        post[r] = 1.0f / (1.0f + expf(-s));
    }
}

// ---------------------------------------------------------------------------
// Hebbian update: outW[o,i] = 2*A*(B*pre[i]*post[o] + C*pre[i] + D*post[o] + E)
// H is (4096,4096,5) interleaved: 4 elements/thread = 20 floats = 5 aligned
// float4 NT loads; one float4 NT store. Pure streaming (960MB in, 192MB out).
// ---------------------------------------------------------------------------
__global__ __launch_bounds__(256) void hebb_delta(
    const float* __restrict__ H, const float* __restrict__ pre,
    const float* __restrict__ post, float* __restrict__ outW)
{
    const size_t t  = (size_t)blockIdx.x * 256 + threadIdx.x;
    const size_t e0 = t * 4;                       // first of 4 elements
    const v4f* h4 = (const v4f*)(H + e0 * 5);

    __builtin_prefetch(H + e0 * 5 + 5120, 0, 1);   // stream ahead (global_prefetch_b8)

    v4f q[5];
    #pragma unroll
    for (int i = 0; i < 5; ++i) q[i] = __builtin_nontemporal_load(h4 + i);
    float f[20];
    #pragma unroll
    for (int i = 0; i < 5; ++i) {
        f[4 * i + 0] = q[i][0]; f[4 * i + 1] = q[i][1];
        f[4 * i + 2] = q[i][2]; f[4 * i + 3] = q[i][3];
    }

    const int o  = (int)(e0 >> 12);
    const int i0 = (int)(e0 & (N4K - 1));
    const float po = post[o];
    const v4f  pr  = *(const v4f*)(pre + i0);

    v4f res;
    #pragma unroll
    for (int j = 0; j < 4; ++j) {
        const float A = f[5 * j + 0], B = f[5 * j + 1], C = f[5 * j + 2],
                    D = f[5 * j + 3], E = f[5 * j + 4];
        const float p = pr[j];
        res[j] = 2.0f * (A * (B * p * po + C * p + D * po + E));
    }
    __builtin_nontemporal_store(res, (v4f*)(outW + e0));
}

// ---------------------------------------------------------------------------
extern "C" void kernel_launch(void* const* d_in, const int* in_sizes, int n_in,
                              void* d_out, int out_size, void* d_ws, size_t ws_size,
                              hipStream_t stream)
{
    const float* x  = (const float*)d_in[0];
    const float* W0 = (const float*)d_in[1];
    const float* b0 = (const float*)d_in[2];
    const float* H0 = (const float*)d_in[3];
    const float* W1 = (const float*)d_in[4];
    const float* b1 = (const float*)d_in[5];
    const float* H1 = (const float*)d_in[6];
    const float* W2 = (const float*)d_in[7];
    const float* b2 = (const float*)d_in[8];
    const float* H2 = (const float*)d_in[9];

    float* out   = (float*)d_out;
    float* y_out = out;                       // 4096
    float* oW0   = out + N4K;                 // 16M
    float* oW1   = oW0 + (size_t)N4K * N4K;
    float* oW2   = oW1 + (size_t)N4K * N4K;

    float* wsf     = (float*)d_ws;
    float* partial = wsf;                     // KSPLIT * 4096
    float* h1      = wsf + KSPLIT * N4K;
    float* h2      = h1 + N4K;
    float* sg      = h2 + N4K;

    const dim3 gGemv(32 * KSPLIT), bGemv(256);
    const dim3 gAct(N4K / 256),    bAct(256);
    const dim3 gDelta((N4K * (size_t)N4K) / (4 * 256)), bDelta(256);

    // layer 0: h1 = tanh(W0 x + b0)
    gemv_partial_f32wmma<<<gGemv, bGemv, 0, stream>>>(W0, x, partial);
    act_bias<<<gAct, bAct, 0, stream>>>(partial, b0, h1, nullptr, 0);
    // layer 1: h2 = tanh(W1 h1 + b1)
    gemv_partial_f32wmma<<<gGemv, bGemv, 0, stream>>>(W1, h1, partial);
    act_bias<<<gAct, bAct, 0, stream>>>(partial, b1, h2, nullptr, 0);
    // layer 2: y = W2 h2 + b2 ; sg = sigmoid(y)
    gemv_partial_f32wmma<<<gGemv, bGemv, 0, stream>>>(W2, h2, partial);
    act_bias<<<gAct, bAct, 0, stream>>>(partial, b2, sg, y_out, 1);

    // Hebbian streams (the bandwidth-dominant 1.15 GB)
    hebb_delta<<<gDelta, bDelta, 0, stream>>>(H0, x,  h1, oW0);
    hebb_delta<<<gDelta, bDelta, 0, stream>>>(H1, h1, h2, oW1);
    hebb_delta<<<gDelta, bDelta, 0, stream>>>(H2, h2, sg, oW2);
}